// INSTA_29515015258436
// MI455X (gfx1250) — compile-verified
//
#include <hip/hip_runtime.h>
#include <hip/hip_bf16.h>

// ---------------------------------------------------------------------------
// INSTA few-shot adaptation for MI455X (gfx1250, wave32).
// All GEMMs on the exact fp32 matrix pipe: V_WMMA_F32_16X16X4_F32.
// Activations kept pixel-major [b][p][c] (p padded 100->112) so every
// A/B fragment is a contiguous b64 load; the K-loop is branch-free and
// software-pipelined (double-buffered fragments, next chunk's loads issued
// before the current chunk's WMMAs -> loads overlap the matrix pipe).
// ---------------------------------------------------------------------------

typedef float v2f __attribute__((ext_vector_type(2)));
typedef float v8f __attribute__((ext_vector_type(8)));

#define CC    640
#define RR    128
#define HW    100
#define HWP   112            // padded pixel rows (7 x 16)
#define K9    9
#define BSUP  25
#define BQRY  75
#define CK9   5760           // C * 9
#define PLANE 64000          // C * HW   (channel-major batch plane)
#define PPAD  (HWP * CC)     // 71680    (pixel-major padded batch plane)

// ---------------------------------------------------------------------------
// Branch-free, software-pipelined fp32 WMMA GEMM.
//   Out[m, n] = sum_k A[m, k] * B(k, n)  (+ bias[n]) (ReLU optional)
// A: row-major, lda; rows padded up to gridDim.y*MT*16.
// B: BT=1 -> B(k,n) = B[n*ldb + k]   (contiguous v2f along k)
//    BT=0 -> B(k,n) = B[k*ldb + n]   (two strided b32 loads)
// GUARD=false: tile is fully stored (padded buffers), no epilogue branches.
// Kdim must be a multiple of 8 and >= 8.
// Fragment layouts per CDNA5 ISA 7.12.2 (wave32):
//   A 16x4: lanes 0-15 rows, v={K0,K1}; lanes 16-31 same rows, v={K2,K3}
//   B 4x16: lanes 0-15 cols, v={K0,K1}; lanes 16-31 cols,      v={K2,K3}
//   C/D  : VGPR r -> M = r (lanes 0-15) / r+8 (lanes 16-31), N = lane&15
// ---------------------------------------------------------------------------
template <int MT, int BT, bool GUARD>
__global__ void gemm_wmma_f32(const float* __restrict__ A,
                              const float* __restrict__ B,
                              const float* __restrict__ bias,
                              float* __restrict__ Out,
                              int N, int Kdim, int lda, int ldb, int ldo,
                              long sA, long sB, long sO,
                              int Mstore, int Nstore, int doRelu)
{
    const int wave = threadIdx.x >> 5;
    const int lane = threadIdx.x & 31;
    const int half = lane >> 4;          // K-pair select: {0,1} vs {2,3}
    const int ln   = lane & 15;

    const int n0 = (blockIdx.x * (blockDim.x >> 5) + wave) * 16;
    if (n0 >= N) return;                 // wave-uniform
    const int m0 = blockIdx.y * (MT * 16);

    const float* Az = A + (long)blockIdx.z * sA;
    const float* Bz = B + (long)blockIdx.z * sB;

    // hoisted per-lane base pointers (K-loop adds only +k)
    const float* Arow[MT];
#pragma unroll
    for (int t = 0; t < MT; ++t)
        Arow[t] = Az + (long)(m0 + t * 16 + ln) * lda + half * 2;

    const float* Bp = BT ? (Bz + (long)(n0 + ln) * ldb + half * 2)
                         : (Bz + (long)(half * 2) * ldb + (n0 + ln));

    __builtin_prefetch(Arow[0], 0, 0);
    __builtin_prefetch(Bp, 0, 0);

    auto loadB = [&](v2f& b, int k) {
        if (BT) {
            b = *(const v2f*)(Bp + k);
        } else {
            b.x = Bp[(long)k * ldb];
            b.y = Bp[(long)k * ldb + ldb];
        }
    };

    v8f acc[MT] = {};
    v2f a0[MT], a1[MT], b0, b1;

    // ---- pipeline prologue: chunk 0 in buf0 ----
    loadB(b0, 0);
#pragma unroll
    for (int t = 0; t < MT; ++t) a0[t] = *(const v2f*)(Arow[t]);

    // ---- steady state: at loop head, buf0 holds chunk k ----
    int k = 0;
    for (; k + 8 < Kdim; k += 8) {
        loadB(b1, k + 4);                              // issue chunk k+4
#pragma unroll
        for (int t = 0; t < MT; ++t) a1[t] = *(const v2f*)(Arow[t] + k + 4);
#pragma unroll
        for (int t = 0; t < MT; ++t)                   // consume chunk k
            acc[t] = __builtin_amdgcn_wmma_f32_16x16x4_f32(
                         false, a0[t], false, b0, (short)0, acc[t], false, false);

        loadB(b0, k + 8);                              // issue chunk k+8
#pragma unroll
        for (int t = 0; t < MT; ++t) a0[t] = *(const v2f*)(Arow[t] + k + 8);
#pragma unroll
        for (int t = 0; t < MT; ++t)                   // consume chunk k+4
            acc[t] = __builtin_amdgcn_wmma_f32_16x16x4_f32(
                         false, a1[t], false, b1, (short)0, acc[t], false, false);
    }

    // ---- tail: k == Kdim-8; buf0 holds chunk k ----
    loadB(b1, k + 4);
#pragma unroll
    for (int t = 0; t < MT; ++t) a1[t] = *(const v2f*)(Arow[t] + k + 4);
#pragma unroll
    for (int t = 0; t < MT; ++t)
        acc[t] = __builtin_amdgcn_wmma_f32_16x16x4_f32(
                     false, a0[t], false, b0, (short)0, acc[t], false, false);
#pragma unroll
    for (int t = 0; t < MT; ++t)
        acc[t] = __builtin_amdgcn_wmma_f32_16x16x4_f32(
                     false, a1[t], false, b1, (short)0, acc[t], false, false);

    // ---- epilogue ----
    float* Oz = Out + (long)blockIdx.z * sO;
    const int col = n0 + ln;
    if (!GUARD) {
        const float bv = bias[col];
#pragma unroll
        for (int t = 0; t < MT; ++t) {
#pragma unroll
            for (int r = 0; r < 8; ++r) {
                const int m = m0 + t * 16 + r + half * 8;
                float v = acc[t][r] + bv;
                if (doRelu && v < 0.f) v = 0.f;
                Oz[(long)m * ldo + col] = v;
            }
        }
    } else {
#pragma unroll
        for (int t = 0; t < MT; ++t) {
#pragma unroll
            for (int r = 0; r < 8; ++r) {
                const int m = m0 + t * 16 + r + half * 8;
                if (m < Mstore && col < Nstore) {
                    float v = acc[t][r] + bias[col];
                    if (doRelu && v < 0.f) v = 0.f;
                    Oz[(long)m * ldo + col] = v;
                }
            }
        }
    }
}

// ---------------------------------------------------------------------------
// Transpose channel-major [b][c][p] -> pixel-major padded [b][pP][c], p>=HW -> 0
// ---------------------------------------------------------------------------
__global__ void transpose_pad(const float* __restrict__ x,
                              float* __restrict__ y, long total)
{
    const long i = (long)blockIdx.x * blockDim.x + threadIdx.x;
    if (i >= total) return;
    const long b   = i / PPAD;
    const int  rem = (int)(i - b * PPAD);
    const int  p   = rem / CC;
    const int  c   = rem - p * CC;
    y[i] = (p < HW) ? x[b * PLANE + (long)c * HW + p] : 0.f;
}

// Zero-pad Ws [9][640] -> [16][640]
__global__ void pad_ws(const float* __restrict__ w, float* __restrict__ y)
{
    const int i = blockIdx.x * blockDim.x + threadIdx.x;
    if (i >= 16 * CC) return;
    const int r = i / CC;
    y[i] = (r < K9) ? w[i] : 0.f;
}

// pooled[b][c] from pixel-major x (rows padded to 32 with zeros)
__global__ void pool_mean_pm(const float* __restrict__ x,
                             float* __restrict__ out)
{
    const int i = blockIdx.x * blockDim.x + threadIdx.x;   // i = b*640 + c
    if (i >= 32 * CC) return;
    const int b = i / CC;
    const int c = i - b * CC;
    if (b >= BSUP) { out[i] = 0.f; return; }
    const float* p = x + (long)b * PPAD + c;
    float s = 0.f;
#pragma unroll
    for (int j = 0; j < HW; ++j) s += p[(long)j * CC];
    out[i] = s * (1.0f / HW);
}

// ---------------------------------------------------------------------------
// task_kernel[c,p,u9] = mean_b( ck_t[b, c*9+u9] * sk_t[b, p*9+u9] )
// sk buffers are already in the raw-reshape linear order p*9+u9.
// ---------------------------------------------------------------------------
__global__ void task_kernel_build(const float* __restrict__ ck,
                                  const float* __restrict__ sk,
                                  float* __restrict__ tk)
{
    const int i = blockIdx.x * blockDim.x + threadIdx.x;   // c*100 + p
    if (i >= CC * HW) return;
    const int c = i / HW;
    const int p = i - c * HW;

    float acc[K9];
#pragma unroll
    for (int u = 0; u < K9; ++u) acc[u] = 0.f;

    for (int b = 0; b < BSUP; ++b) {
        const float* ckb = ck + (long)b * CK9 + c * K9;
        const float* skb = sk + (long)b * (K9 * HW) + p * K9;
#pragma unroll
        for (int u = 0; u < K9; ++u) acc[u] += ckb[u] * skb[u];
    }
#pragma unroll
    for (int u = 0; u < K9; ++u)
        tk[(long)i * K9 + u] = acc[u] * (1.0f / BSUP);
}

// adapted_query[b,c,p] = sum_{u,v} tk[c,p,u9] * q[b,c, i+u-1, j+v-1]
__global__ void invol_query(const float* __restrict__ x,
                            const float* __restrict__ tk,
                            float* __restrict__ out, int total)
{
    const int i = blockIdx.x * blockDim.x + threadIdx.x;
    if (i >= total) return;
    const int p  = i % HW;
    const int bc = i / HW;
    const int c  = bc % CC;
    const int ii = p / 10, jj = p % 10;

    const float* xb = x + (long)bc * HW;
    const float* t  = tk + ((long)c * HW + p) * K9;

    float acc = 0.f;
#pragma unroll
    for (int u = 0; u < 3; ++u) {
        const int y = ii + u - 1;
        if ((unsigned)y < 10u) {
#pragma unroll
            for (int v = 0; v < 3; ++v) {
                const int xc = jj + v - 1;
                if ((unsigned)xc < 10u)
                    acc += t[u * 3 + v] * xb[y * 10 + xc];
            }
        }
    }
    out[i] = acc;
}

// adapted_support: fused = ck_s*sk_s*tk consumed on the fly (57 MB avoided)
__global__ void invol_support(const float* __restrict__ x,
                              const float* __restrict__ tk,
                              const float* __restrict__ ck,
                              const float* __restrict__ sk,
                              float* __restrict__ out, int total)
{
    const int i = blockIdx.x * blockDim.x + threadIdx.x;
    if (i >= total) return;
    const int p  = i % HW;
    const int bc = i / HW;
    const int c  = bc % CC;
    const int b  = bc / CC;
    const int ii = p / 10, jj = p % 10;

    const float* xb  = x  + (long)bc * HW;
    const float* t   = tk + ((long)c * HW + p) * K9;
    const float* ckb = ck + (long)b * CK9 + c * K9;
    const float* skb = sk + (long)b * (K9 * HW) + p * K9;

    float acc = 0.f;
#pragma unroll
    for (int u = 0; u < 3; ++u) {
        const int y = ii + u - 1;
        if ((unsigned)y < 10u) {
#pragma unroll
            for (int v = 0; v < 3; ++v) {
                const int xc = jj + v - 1;
                if ((unsigned)xc < 10u) {
                    const int u9 = u * 3 + v;
                    acc += ckb[u9] * skb[u9] * t[u9] * xb[y * 10 + xc];
                }
            }
        }
    }
    out[i] = acc;
}

// ---------------------------------------------------------------------------
extern "C" void kernel_launch(void* const* d_in, const int* in_sizes, int n_in,
                              void* d_out, int out_size, void* d_ws, size_t ws_size,
                              hipStream_t stream)
{
    const float* support = (const float*)d_in[0];
    const float* query   = (const float*)d_in[1];
    const float* Wc[4]   = {(const float*)d_in[2], (const float*)d_in[4],
                            (const float*)d_in[6], (const float*)d_in[8]};
    const float* bc[4]   = {(const float*)d_in[3], (const float*)d_in[5],
                            (const float*)d_in[7], (const float*)d_in[9]};
    const float* W1 = (const float*)d_in[10];
    const float* b1 = (const float*)d_in[11];
    const float* W2 = (const float*)d_in[12];
    const float* b2 = (const float*)d_in[13];
    const float* Ws = (const float*)d_in[14];
    const float* bs = (const float*)d_in[15];

    // ---- workspace carve-up (floats) ----
    float* ws       = (float*)d_ws;
    float* St       = ws;                        // 25*71680 = 1,792,000
    float* tmpA     = St       + BSUP * PPAD;    // 1,792,000
    float* tmpB     = tmpA     + BSUP * PPAD;    // 1,792,000
    float* WsPad    = tmpB     + BSUP * PPAD;    // 10,240
    float* pooled_t = WsPad    + 16 * CC;        // 20,480 (32 x 640)
    float* pooled_s = pooled_t + 32 * CC;        // 20,480
    float* hid_t    = pooled_s + 32 * CC;        // 4,096  (32 x 128)
    float* hid_s    = hid_t    + 32 * RR;        // 4,096
    float* ck_t     = hid_s    + 32 * RR;        // 184,320 (32 x 5760)
    float* ck_s     = ck_t     + 32 * CK9;       // 184,320
    float* sk_t     = ck_s     + 32 * CK9;       // 22,500  (25 x 900)
    float* sk_s     = sk_t     + BSUP * K9 * HW; // 22,500
    float* tk       = sk_s     + BSUP * K9 * HW; // 576,000

    float* outS = (float*)d_out;                 // adapted_support [25,640,10,10]
    float* outQ = outS + BSUP * PLANE;           // adapted_query   [75,640,10,10]

    const dim3 blk(256);   // 8 waves (wave32)

    // ---- 0) layout prep ----
    transpose_pad<<<(BSUP * PPAD + 255) / 256, 256, 0, stream>>>(
        support, St, (long)BSUP * PPAD);
    pad_ws<<<(16 * CC + 255) / 256, 256, 0, stream>>>(Ws, WsPad);

    // ---- 1) ContextLearningModule: 4 chained 1x1 convs ----
    // Out[p,o] = Y[p,c] * Wc[o,c]  (+bc[o]); M=112(pad), N=640, K=640 per batch.
    // MT=7 -> one wave covers all 112 rows of one 16-col tile; 40 N-tiles.
    {
        const float* src = St;
        float* dsts[4] = {tmpA, tmpB, tmpA, tmpB};
        for (int l = 0; l < 4; ++l) {
            gemm_wmma_f32<7, 1, false><<<dim3(5, 1, BSUP), blk, 0, stream>>>(
                src, Wc[l], bc[l], dsts[l],
                CC, CC,            /* N K          */
                CC, CC, CC,        /* lda ldbT ldo */
                PPAD, 0, PPAD,     /* strides A B O */
                HWP, CC, 0);       /* Mstore Nstore relu */
            src = dsts[l];
        }
    }
    const float* tsr = tmpB;   // conv-chain output (pixel-major, padded)

    // ---- 2) Global average pools (padded to 32 rows with zeros) ----
    pool_mean_pm<<<(32 * CC + 255) / 256, 256, 0, stream>>>(tsr, pooled_t);
    pool_mean_pm<<<(32 * CC + 255) / 256, 256, 0, stream>>>(St, pooled_s);

    // ---- 3) Channel-branch MLP ----
    // GEMM1: hid = relu(pooled @ W1 + b1); M=32(pad), N=128, K=640
    gemm_wmma_f32<2, 0, false><<<dim3(1, 1, 1), blk, 0, stream>>>(
        pooled_t, W1, b1, hid_t, RR, CC, CC, RR, RR, 0, 0, 0, 32, RR, 1);
    gemm_wmma_f32<2, 0, false><<<dim3(1, 1, 1), blk, 0, stream>>>(
        pooled_s, W1, b1, hid_s, RR, CC, CC, RR, RR, 0, 0, 0, 32, RR, 1);
    // GEMM2: ck = hid @ W2 + b2; M=32(pad), N=5760, K=128
    gemm_wmma_f32<2, 0, false><<<dim3(45, 1, 1), blk, 0, stream>>>(
        hid_t, W2, b2, ck_t, CK9, RR, RR, CK9, CK9, 0, 0, 0, 32, CK9, 0);
    gemm_wmma_f32<2, 0, false><<<dim3(45, 1, 1), blk, 0, stream>>>(
        hid_s, W2, b2, ck_s, CK9, RR, RR, CK9, CK9, 0, 0, 0, 32, CK9, 0);

    // ---- 4) Spatial branch: sk[p, u9] = Y[p,c] * Ws[u9,c] + bs[u9] ----
    // ldo=9 puts the output directly in the raw-reshape order p*9+u9.
    gemm_wmma_f32<7, 1, true><<<dim3(1, 1, BSUP), blk, 0, stream>>>(
        tsr, WsPad, bs, sk_t, 16, CC, CC, CC, K9, PPAD, 0, K9 * HW, HW, K9, 0);
    gemm_wmma_f32<7, 1, true><<<dim3(1, 1, BSUP), blk, 0, stream>>>(
        St, WsPad, bs, sk_s, 16, CC, CC, CC, K9, PPAD, 0, K9 * HW, HW, K9, 0);

    // ---- 5) Task kernel: mean_b(ck_t * sk_t) -> [640,100,9] ----
    task_kernel_build<<<(CC * HW + 255) / 256, 256, 0, stream>>>(ck_t, sk_t, tk);

    // ---- 6) Involutions (consume original channel-major inputs) ----
    invol_support<<<(BSUP * PLANE + 255) / 256, 256, 0, stream>>>(
        support, tk, ck_s, sk_s, outS, BSUP * PLANE);
    invol_query<<<(BQRY * PLANE + 255) / 256, 256, 0, stream>>>(
        query, tk, outQ, BQRY * PLANE);
}